// FourierConv2D_17325898072688
// MI455X (gfx1250) — compile-verified
//
#include <hip/hip_runtime.h>
#include <hip/hip_bf16.h>
#include <stdint.h>

// FourierConv2D for MI455X (gfx1250), wave32.
// B=16, H=W=128, C=16, F=32, FFT 256x256, rfft W-axis -> 129 bins.
// Stages B, C, D run on the matrix pipe (v_wmma_f32_16x16x4_f32).

typedef float v2f __attribute__((ext_vector_type(2)));
typedef float v8f __attribute__((ext_vector_type(8)));

#define TWOPI_256 0.0245436926061702596f   // 2*pi/256

static constexpr int Bn = 16, Hn = 128, Wn = 128, Cn = 16, Fn = 32;
static constexpr int KH = 256;   // full freq rows
static constexpr int KW = 129;   // rfft cols

static __device__ inline v8f wmma4(v2f a, v2f b, v8f c) {
    return __builtin_amdgcn_wmma_f32_16x16x4_f32(false, a, false, b, (short)0, c,
                                                 false, false);
}

// Complex GEMM strip: D(16x16) += E(16xK) * X(Kx16), twiddle E in LDS [k][m],
// X columns in global memory at colbase + k*rowStride.  K = 4*nchunks.
static __device__ inline void cgemm_strip(const float2* __restrict__ colbase,
                                          long rowStride,
                                          const float2* __restrict__ strip,
                                          int lane, int nchunks,
                                          v8f& outRe, v8f& outIm) {
    const int m = lane & 15;      // A row / B col / D col
    const int kgrp = lane >> 4;   // k-pair selector
    v8f accRR = {}, accII = {}, accIX = {};
#pragma unroll 4
    for (int kk = 0; kk < nchunks; ++kk) {
        const int k0 = kk * 4 + kgrp * 2;
        float2 e0 = strip[k0 * 16 + m];
        float2 e1 = strip[(k0 + 1) * 16 + m];
        v2f ar, ai;
        ar[0] = e0.x; ar[1] = e1.x;
        ai[0] = e0.y; ai[1] = e1.y;
        float2 x0 = colbase[(long)k0 * rowStride];
        float2 x1 = colbase[(long)(k0 + 1) * rowStride];
        v2f br, bi;
        br[0] = x0.x; br[1] = x1.x;
        bi[0] = x0.y; bi[1] = x1.y;
        accRR = wmma4(ar, br, accRR);
        accII = wmma4(ai, bi, accII);
        accIX = wmma4(ar, bi, accIX);
        accIX = wmma4(ai, br, accIX);
    }
    outRe = accRR - accII;
    outIm = accIX;
}

// ---------------- Stage A: row-wise rfft (W axis), rows h<128 only -------------
// A1[b][h][w'][c] = sum_{n<128} img[b,h,n,c] * e^{-2pi i w' n /256}
__global__ __launch_bounds__(256) void stageA_row_rfft(const float* __restrict__ img,
                                                       float2* __restrict__ A1) {
    __shared__ float2 tw[256];                         // unit circle e^{+i 2pi p/256}
    {
        float s, c;
        __sincosf((float)threadIdx.x * TWOPI_256, &s, &c);
        tw[threadIdx.x] = make_float2(c, s);
    }
    __syncthreads();

    long tid = (long)blockIdx.x * 256 + threadIdx.x;   // 16*128*129*16 = 4,227,072
    int c = (int)(tid & 15);
    long t = tid >> 4;
    int w = (int)(t % KW); t /= KW;
    int h = (int)(t % Hn);
    int b = (int)(t / Hn);
    const float* base = img + (((long)b * Hn + h) * Wn) * Cn + c;
    float sr = 0.f, si = 0.f;
    for (int n = 0; n < Wn; ++n) {
        float v = base[(long)n * Cn];
        float2 e = tw[(w * n) & 255];                  // broadcast LDS read
        sr = fmaf(v, e.x, sr);                         // e^{-i theta}: (cos, -sin)
        si = fmaf(-v, e.y, si);
    }
    A1[tid] = make_float2(sr, si);
}

// ---------------- Stage B: forward FFT along H as WMMA complex GEMM ------------
// Fimg[b][k][w'][c] = sum_{n<128} E[k,n] * A1[b,n,w',c],  E = e^{-2pi i k n/256}
// Block: 8 waves share one 16-row twiddle strip (16 KB LDS); 32 units/block.
__global__ __launch_bounds__(256) void stageB_col_fft_wmma(const float2* __restrict__ A1,
                                                           float2* __restrict__ Fimg) {
    __shared__ float2 strip[128 * 16];                 // [n][m] for 16 k-rows
    const int mtile = blockIdx.x & 15;                 // k-tile: rows mtile*16..+15
    const int grp   = blockIdx.x >> 4;                 // 65 unit-groups of 32
    for (int i = threadIdx.x; i < 128 * 16; i += 256) {
        int n = i >> 4, m = i & 15;
        int p = (((mtile * 16 + m) * n) & 255);
        float s, c;
        __sincosf((float)p * TWOPI_256, &s, &c);
        strip[i] = make_float2(c, -s);                 // e^{-i theta}
    }
    __syncthreads();

    const int lane = threadIdx.x & 31;
    const int wave = threadIdx.x >> 5;
    const int col  = lane & 15;                        // c
#pragma unroll
    for (int i = 0; i < 4; ++i) {
        int u = grp * 32 + wave * 4 + i;               // unit = (b, w')
        if (u >= Bn * KW) break;                       // 2064 units per mtile
        int b = u / KW;
        int w = u - b * KW;
        const float2* colbase =
            A1 + ((long)b * Hn * KW + w) * Cn + col;   // + n * (KW*Cn)
        v8f re, im;
        cgemm_strip(colbase, (long)KW * Cn, strip, lane, 32, re, im);
        const int kgrp = lane >> 4;
#pragma unroll
        for (int r = 0; r < 8; ++r) {
            int k = mtile * 16 + r + kgrp * 8;
            Fimg[(((long)b * KH + k) * KW + w) * Cn + col] = make_float2(re[r], im[r]);
        }
    }
}

// ---------------- Stage C: per-bin complex GEMM via WMMA f32 16x16x4 -----------
// out_f[b][k][w'][f] = sum_c Fimg[b,k,w',c] * (kr + i ki)[k,w',f,c]
__global__ __launch_bounds__(256) void stageC_wmma_cgemm(const float2* __restrict__ Fimg,
                                                         const float*  __restrict__ kr,
                                                         const float*  __restrict__ ki,
                                                         float2* __restrict__ outf) {
    const int lane = threadIdx.x & 31;
    const int wave = threadIdx.x >> 5;
    const long unit = (long)blockIdx.x * 8 + wave;     // 66048 units (exact)
    const int bin = (int)(unit >> 1);
    const int nt  = (int)(unit & 1);
    const int h = bin / KW;
    const int w = bin - h * KW;
    const int row  = lane & 15;
    const int kgrp = lane >> 4;

    const float2* abase = Fimg + ((((long)row * KH + h) * KW + w) * Cn);
    v2f ar[4], ai[4];
#pragma unroll
    for (int kk = 0; kk < 4; ++kk) {
        int c0 = kk * 4 + kgrp * 2;
        float2 e0 = abase[c0];
        float2 e1 = abase[c0 + 1];
        ar[kk][0] = e0.x; ar[kk][1] = e1.x;
        ai[kk][0] = e0.y; ai[kk][1] = e1.y;
    }
    const long kofs = (((long)h * KW + w) * Fn + (nt * 16 + row)) * Cn;
    const float2* krb = (const float2*)(kr + kofs);
    const float2* kib = (const float2*)(ki + kofs);
    __builtin_prefetch(kr + kofs + (long)Fn * Cn, 0, 1);
    v2f br[4], bi[4];
#pragma unroll
    for (int kk = 0; kk < 4; ++kk) {
        int cp = (kk * 4 + kgrp * 2) >> 1;
        float2 r = krb[cp];
        float2 im = kib[cp];
        br[kk][0] = r.x;  br[kk][1] = r.y;
        bi[kk][0] = im.x; bi[kk][1] = im.y;
    }
    v8f accRR = {}, accII = {}, accIm = {};
#pragma unroll
    for (int kk = 0; kk < 4; ++kk) accRR = wmma4(ar[kk], br[kk], accRR);
#pragma unroll
    for (int kk = 0; kk < 4; ++kk) accII = wmma4(ai[kk], bi[kk], accII);
#pragma unroll
    for (int kk = 0; kk < 4; ++kk) accIm = wmma4(ar[kk], bi[kk], accIm);
#pragma unroll
    for (int kk = 0; kk < 4; ++kk) accIm = wmma4(ai[kk], br[kk], accIm);
    v8f accRe = accRR - accII;
#pragma unroll
    for (int r = 0; r < 8; ++r) {
        int m = r + kgrp * 8;                          // batch b
        long o = ((((long)m * KH + h) * KW + w) * Fn) + nt * 16 + row;
        outf[o] = make_float2(accRe[r], accIm[r]);
    }
}

// ---------------- Stage D: inverse FFT along H as WMMA GEMM, rows 63..190 ------
// T[b][ho][w'][f] = sum_{k<256} E[ho,k]*out_f[b,k,w',f], E = e^{+2pi i (ho+63)k/256}
__global__ __launch_bounds__(256) void stageD_col_ifft_wmma(const float2* __restrict__ outf,
                                                            float2* __restrict__ T) {
    __shared__ float2 strip[256 * 16];                 // [k][m], 32 KB
    const int mtile = blockIdx.x & 7;                  // ho rows mtile*16..+15
    const int grp   = blockIdx.x >> 3;                 // 129 unit-groups of 32
    for (int i = threadIdx.x; i < 256 * 16; i += 256) {
        int k = i >> 4, m = i & 15;
        int p = (((mtile * 16 + m + 63) * k) & 255);
        float s, c;
        __sincosf((float)p * TWOPI_256, &s, &c);
        strip[i] = make_float2(c, s);                  // e^{+i theta}
    }
    __syncthreads();

    const int lane = threadIdx.x & 31;
    const int wave = threadIdx.x >> 5;
    const int col  = lane & 15;                        // f within tile
#pragma unroll
    for (int i = 0; i < 4; ++i) {
        int u = grp * 32 + wave * 4 + i;               // unit = (b, w', ftile); 4128 total
        int ft = u & 1;
        int rest = u >> 1;
        int b = rest / KW;
        int w = rest - b * KW;
        const float2* colbase =
            outf + ((long)b * KH * KW + w) * Fn + ft * 16 + col;  // + k * (KW*Fn)
        v8f re, im;
        cgemm_strip(colbase, (long)KW * Fn, strip, lane, 64, re, im);
        const int kgrp = lane >> 4;
#pragma unroll
        for (int r = 0; r < 8; ++r) {
            int ho = mtile * 16 + r + kgrp * 8;
            T[(((long)b * Hn + ho) * KW + w) * Fn + ft * 16 + col] =
                make_float2(re[r], im[r]);
        }
    }
}

// ---------------- Stage E: irfft along W (hermitian), crop cols, + bias --------
__global__ __launch_bounds__(256) void stageE_row_irfft(const float2* __restrict__ T,
                                                        const float* __restrict__ bias,
                                                        float* __restrict__ out) {
    __shared__ float2 tw[256];
    {
        float s, c;
        __sincosf((float)threadIdx.x * TWOPI_256, &s, &c);
        tw[threadIdx.x] = make_float2(c, s);
    }
    __syncthreads();

    long tid = (long)blockIdx.x * 256 + threadIdx.x;   // 16*128*128*32 = 8,388,608
    int f = (int)(tid & 31);
    long t = tid >> 5;
    int wo = (int)(t % Wn); t /= Wn;
    int ho = (int)(t % Hn);
    int b = (int)(t / Hn);
    const int ww = wo + 63;
    const float2* base = T + ((((long)b * Hn + ho) * KW) * Fn) + f;
    float2 x0   = base[0];
    float2 x128 = base[(long)128 * Fn];
    float acc = x0.x + ((ww & 1) ? -x128.x : x128.x);
    for (int k = 1; k < 128; ++k) {
        float2 x = base[(long)k * Fn];
        float2 e = tw[(k * ww) & 255];                 // broadcast LDS read
        acc += 2.f * (x.x * e.x - x.y * e.y);          // Re(X * e^{+i theta})
    }
    out[tid] = acc * (1.f / 65536.f) + bias[f];
}

extern "C" void kernel_launch(void* const* d_in, const int* in_sizes, int n_in,
                              void* d_out, int out_size, void* d_ws, size_t ws_size,
                              hipStream_t stream) {
    const float* img  = (const float*)d_in[0];
    const float* kr   = (const float*)d_in[1];
    const float* ki   = (const float*)d_in[2];
    const float* bias = (const float*)d_in[3];
    float* out = (float*)d_out;

    // workspace layout (bytes), with reuse: T overlaps dead A1+Fimg region
    char* ws = (char*)d_ws;
    const size_t szA1   = (size_t)Bn * Hn * KW * Cn * sizeof(float2);   //  33.8 MB
    const size_t szFimg = (size_t)Bn * KH * KW * Cn * sizeof(float2);   //  67.6 MB
    float2* A1   = (float2*)(ws);
    float2* Fimg = (float2*)(ws + szA1);
    float2* outf = (float2*)(ws + szA1 + szFimg);                       // 135.3 MB
    float2* T    = (float2*)(ws);                                       // reuses A1+Fimg

    stageA_row_rfft    <<<16512, 256, 0, stream>>>(img, A1);
    stageB_col_fft_wmma<<<16 * 65, 256, 0, stream>>>(A1, Fimg);         // 1040 blocks
    stageC_wmma_cgemm  <<<8256, 256, 0, stream>>>(Fimg, kr, ki, outf);
    stageD_col_ifft_wmma<<<8 * 129, 256, 0, stream>>>(outf, T);         // 1032 blocks
    stageE_row_irfft   <<<32768, 256, 0, stream>>>(T, bias, out);
    (void)in_sizes; (void)n_in; (void)out_size; (void)ws_size;
}